// SimpleRNN_torch_31516470018370
// MI455X (gfx1250) — compile-verified
//
#include <hip/hip_runtime.h>
#include <cstdint>
#include <cstddef>

// ---------------------------------------------------------------------------
// EMA scan  s_t = c*x_t + (1-c)*s_{t-1},  c = 0.04
// inputs:  x[64][16000][40] f32, s0[64][40] f32
// output:  s[64][16000][40] f32
//
// 3-pass chunked parallel scan; pass 3 does the intra-block (16-step) scan as
// a lower-triangular 16x16 matmul on the WMMA pipe (V_WMMA_F32_16X16X4_F32 x4).
// Pass 3 is split into a branch-free full-tile kernel (units 0..31) and a
// guarded partial-tile kernel (units 32..39) so the common case has no EXEC
// manipulation in the hot loop (WMMA requires EXEC all-ones anyway).
// ---------------------------------------------------------------------------

#define SEQ    16000
#define UNITS  40
#define BATCH  64
#define CHUNK  400              // timesteps per chunk
#define NCHUNK (SEQ / CHUNK)    // 40
#define BLK    16               // WMMA block of timesteps
#define NBLK   (CHUNK / BLK)    // 25
#define FULLT  2                // full 16-unit tiles (units 0..31)

#define CCOEF 0.04f
#define QCOEF 0.96f

typedef float v2f __attribute__((ext_vector_type(2)));
typedef float v8f __attribute__((ext_vector_type(8)));

__host__ __device__ constexpr float qpow_const(int n) {
    float r = 1.0f;
    for (int i = 0; i < n; ++i) r *= QCOEF;
    return r;
}

__device__ __forceinline__ float qpow_rt(int e) {
    float r = 1.0f;
    for (int i = 0; i < e; ++i) r *= QCOEF;
    return r;
}

// ---------------------------------------------------------------------------
// Pass 1: per-(batch, chunk, unit) carry with zero initial state.
// ---------------------------------------------------------------------------
__global__ void __launch_bounds__(64)
ema_chunk_carry(const float* __restrict__ x, float* __restrict__ carryB) {
    const int b = blockIdx.x;   // 0..63
    const int j = blockIdx.y;   // 0..39
    const int u = threadIdx.x;  // 0..63, active < 40
    if (u >= UNITS) return;

    const float* p = x + ((size_t)b * SEQ + (size_t)j * CHUNK) * UNITS + u;
    float s = 0.0f;
#pragma unroll 8
    for (int k = 0; k < CHUNK; ++k) {
        s = fmaf(QCOEF, s, CCOEF * p[(size_t)k * UNITS]);
    }
    carryB[((size_t)b * NCHUNK + j) * UNITS + u] = s;
}

// ---------------------------------------------------------------------------
// Pass 2: sequential combine of chunk carries -> per-chunk start states.
// ---------------------------------------------------------------------------
__global__ void __launch_bounds__(256)
ema_chunk_prefix(const float* __restrict__ s0,
                 const float* __restrict__ carryB,
                 float* __restrict__ sstart) {
    const int tid = blockIdx.x * blockDim.x + threadIdx.x;
    if (tid >= BATCH * UNITS) return;
    const int b = tid / UNITS;
    const int u = tid % UNITS;

    constexpr float ACH = qpow_const(CHUNK);  // 0.96^400 ~ 8e-8
    float s = s0[(size_t)b * UNITS + u];
#pragma unroll
    for (int j = 0; j < NCHUNK; ++j) {
        const size_t idx = ((size_t)b * NCHUNK + j) * UNITS + u;
        sstart[idx] = s;
        s = fmaf(ACH, s, carryB[idx]);
    }
}

// ---------------------------------------------------------------------------
// Pass 3 body: one wave per (batch, unit-tile, chunk).
// 16-step block:  S = L @ X + d (x) carry, with L[t,k] = c*q^(t-k) lower-tri.
// L@X via 4 chained V_WMMA_F32_16X16X4_F32.
//
// f32 16x4 A layout (ISA 7.12.2): lanes 0-15: M=lane, vgpr0=K0, vgpr1=K1;
//                                 lanes 16-31: M=lane-16, vgpr0=K2, vgpr1=K3.
// f32 4x16 B layout (mirrors A/C striping): lane holds column N=lane&15;
//                                 vgpr0 = row K=2*hi, vgpr1 = row K=2*hi+1.
// f32 16x16 C/D layout: vgpr r = row M = r + 8*hi, column N = lane&15.
// ---------------------------------------------------------------------------
template <bool FULL>
__device__ __forceinline__ void
ema_wmma_scan_body(const float* __restrict__ x,
                   const float* __restrict__ sstart,
                   float* __restrict__ out,
                   int b, int utile, int j) {
    const int lane = threadIdx.x & 31;
    const int n    = lane & 15;        // column/unit within tile; M for A rows
    const int hi   = lane >> 4;        // lane half

    const int  u      = utile * 16 + n;
    const bool uvalid = FULL || (u < UNITS);
    const int  uc     = uvalid ? u : 0;

    // --- constant A operands: four 16x4 column-slices of L ------------------
    v2f A[4];
#pragma unroll
    for (int jj = 0; jj < 4; ++jj) {
        const int k0 = jj * 4 + hi * 2;
        A[jj].x = (k0     <= n) ? CCOEF * qpow_rt(n - k0)     : 0.0f;
        A[jj].y = (k0 + 1 <= n) ? CCOEF * qpow_rt(n - k0 - 1) : 0.0f;
    }

    // --- per-row decay factors d[t] = q^(t+1), t = r + 8*hi -----------------
    float dp[8];
    {
        float v = qpow_rt(8 * hi + 1);
#pragma unroll
        for (int r = 0; r < 8; ++r) { dp[r] = v; v *= QCOEF; }
    }

    // --- initial carry for this chunk --------------------------------------
    float carry = sstart[((size_t)b * NCHUNK + j) * UNITS + uc];
    if (!FULL) carry = uvalid ? carry : 0.0f;

    const float* xb = x   + ((size_t)b * SEQ + (size_t)j * CHUNK) * UNITS;
    float*       ob = out + ((size_t)b * SEQ + (size_t)j * CHUNK) * UNITS;

    for (int blk = 0; blk < NBLK; ++blk) {
        const int t0 = blk * BLK;

        // prefetch next block's rows into cache (global_prefetch_b8)
        if (blk + 1 < NBLK) {
            __builtin_prefetch(xb + (size_t)(t0 + BLK + hi * 8) * UNITS + uc, 0, 0);
        }

        // --- load B operands: four 4x16 time-slices of X --------------------
        v2f Bv[4];
#pragma unroll
        for (int jj = 0; jj < 4; ++jj) {
            const int ta = t0 + jj * 4 + hi * 2;
            float bx = xb[(size_t)ta       * UNITS + uc];
            float by = xb[(size_t)(ta + 1) * UNITS + uc];
            Bv[jj].x = (FULL || uvalid) ? bx : 0.0f;
            Bv[jj].y = (FULL || uvalid) ? by : 0.0f;
        }

        // --- D = L @ X via chained fp32 WMMA --------------------------------
        v8f acc = {};
        acc = __builtin_amdgcn_wmma_f32_16x16x4_f32(false, A[0], false, Bv[0],
                                                    (short)0, acc, false, false);
        acc = __builtin_amdgcn_wmma_f32_16x16x4_f32(false, A[1], false, Bv[1],
                                                    (short)0, acc, false, false);
        acc = __builtin_amdgcn_wmma_f32_16x16x4_f32(false, A[2], false, Bv[2],
                                                    (short)0, acc, false, false);
        acc = __builtin_amdgcn_wmma_f32_16x16x4_f32(false, A[3], false, Bv[3],
                                                    (short)0, acc, false, false);

        // --- add carry term and store --------------------------------------
        float last = 0.0f;
#pragma unroll
        for (int r = 0; r < 8; ++r) {
            const float v = fmaf(dp[r], carry, acc[r]);
            const int   t = r + 8 * hi;
            if (FULL || uvalid) ob[(size_t)(t0 + t) * UNITS + u] = v;
            if (r == 7) last = v;  // hi-half r=7 is row M=15
        }

        // new carry[n] = S[15][n], held by lane 16+n
        carry = __shfl(last, 16 + n, 32);
    }
}

// Full tiles: utile in {0,1}, all 16 units valid -> branch-free hot loop.
__global__ void __launch_bounds__(32)
ema_wmma_scan_full(const float* __restrict__ x,
                   const float* __restrict__ sstart,
                   float* __restrict__ out) {
    const int gid   = blockIdx.x;           // 0 .. BATCH*FULLT*NCHUNK-1
    const int b     = gid / (FULLT * NCHUNK);
    const int rem   = gid % (FULLT * NCHUNK);
    const int utile = rem / NCHUNK;
    const int j     = rem % NCHUNK;
    ema_wmma_scan_body<true>(x, sstart, out, b, utile, j);
}

// Partial tile: utile == 2, units 32..39 valid, 40..47 masked.
__global__ void __launch_bounds__(32)
ema_wmma_scan_part(const float* __restrict__ x,
                   const float* __restrict__ sstart,
                   float* __restrict__ out) {
    const int gid = blockIdx.x;             // 0 .. BATCH*NCHUNK-1
    const int b   = gid / NCHUNK;
    const int j   = gid % NCHUNK;
    ema_wmma_scan_body<false>(x, sstart, out, b, /*utile=*/FULLT, j);
}

// ---------------------------------------------------------------------------
extern "C" void kernel_launch(void* const* d_in, const int* in_sizes, int n_in,
                              void* d_out, int out_size, void* d_ws, size_t ws_size,
                              hipStream_t stream) {
    const float* x  = (const float*)d_in[0];  // (64, 16000, 40)
    const float* s0 = (const float*)d_in[1];  // (64, 40)
    float* out = (float*)d_out;

    float* carryB = (float*)d_ws;                                  // 64*40*40 f32
    float* sstart = carryB + (size_t)BATCH * NCHUNK * UNITS;       // 64*40*40 f32

    dim3 g1(BATCH, NCHUNK);
    ema_chunk_carry<<<g1, 64, 0, stream>>>(x, carryB);

    ema_chunk_prefix<<<(BATCH * UNITS + 255) / 256, 256, 0, stream>>>(s0, carryB, sstart);

    ema_wmma_scan_full<<<BATCH * FULLT * NCHUNK, 32, 0, stream>>>(x, sstart, out);
    ema_wmma_scan_part<<<BATCH * NCHUNK, 32, 0, stream>>>(x, sstart, out);
}